// ShortConvolution_5016521802520
// MI455X (gfx1250) — compile-verified
//
#include <hip/hip_runtime.h>

// Causal depthwise conv (W=4 taps over T) + SiLU, fp32.
// x (B=8, T=4096, D=2048), kernel (4, 1, 2048) -> y (B, T, D).
//
// HBM-roofline streaming kernel (~537 MB total => ~23 us @ 23.3 TB/s):
//  * each thread owns one float4 channel group and slides a 3-deep register
//    window over TC consecutive time steps -> every x element read once
//    (+3-row halo per chunk), every y element written once.
//  * non-temporal B128 loads/stores keep the one-shot stream out of L2.
//  * global_prefetch_b8 covers the 8 KB stride down T.
//  * block-shared weight slice (16 KB) is staged via the CDNA5 async path
//    (global_load_async_to_lds_b128 + s_wait_asynccnt), then read back with
//    ds_load_b128. Each lane reads only what it staged -> no barrier needed.

typedef float v4f __attribute__((ext_vector_type(4)));
typedef __attribute__((address_space(3))) float lds_float;

constexpr int B   = 8;
constexpr int T   = 4096;
constexpr int D   = 2048;
constexpr int D4  = D / 4;  // 512 float4 columns
constexpr int TC  = 64;     // time steps per thread (8192 wave32s total)
constexpr int BLK = 256;    // 8 wave32s per block

__device__ __forceinline__ v4f silu4(v4f v) {
  v4f r;
#pragma unroll
  for (int i = 0; i < 4; ++i) {
    float a = v[i];
    // exp(-a) = exp2(-a*log2 e); v_exp_f32/v_rcp_f32 are TRANS ops that
    // co-execute with the FMA stream. Saturates cleanly for large |a|.
    float e = __builtin_amdgcn_exp2f(a * -1.44269504088896340736f);
    r[i] = a * __builtin_amdgcn_rcpf(1.0f + e);
  }
  return r;
}

__global__ __launch_bounds__(BLK) void dwconv_silu_kernel(
    const float* __restrict__ xg, const float* __restrict__ wg,
    float* __restrict__ yg) {
  __shared__ float sw[4][BLK * 4];  // 16 KB: this block's weight slice

  const int d4 = blockIdx.x * BLK + threadIdx.x;  // 0 .. D4-1
  const int t0 = blockIdx.y * TC;                 // chunk start in T
  const int b  = blockIdx.z;

  const v4f* __restrict__ x4 = (const v4f*)xg;
  v4f* __restrict__ y4 = (v4f*)yg;

  // Stage the 4 per-channel taps for this lane's float4 into LDS via the
  // gfx1250 async-tensor path (ASYNCcnt-tracked, no VGPR destinations).
#pragma unroll
  for (int k = 0; k < 4; ++k) {
    const float* ga = wg + k * D + 4 * d4;
    unsigned la = (unsigned)(size_t)(lds_float*)&sw[k][4 * threadIdx.x];
    asm volatile("global_load_async_to_lds_b128 %0, %1, off"
                 :: "v"(la), "v"(ga) : "memory");
  }

  size_t idx = ((size_t)b * T + t0) * (size_t)D4 + (size_t)d4;

  // Sliding window: h0 = x[t-3], h1 = x[t-2], h2 = x[t-1].
  v4f h0 = {0.f, 0.f, 0.f, 0.f};
  v4f h1 = {0.f, 0.f, 0.f, 0.f};
  v4f h2 = {0.f, 0.f, 0.f, 0.f};
  if (t0 != 0) {  // uniform per block; halo stays inside the same batch row
    h0 = __builtin_nontemporal_load(&x4[idx - 3 * (size_t)D4]);
    h1 = __builtin_nontemporal_load(&x4[idx - 2 * (size_t)D4]);
    h2 = __builtin_nontemporal_load(&x4[idx - 1 * (size_t)D4]);
  }

  // Taps: k pairs with x[t + k - 3] (k=3 is "current").
  asm volatile("s_wait_asynccnt 0x0" ::: "memory");
  const v4f w0 = *(const v4f*)&sw[0][4 * threadIdx.x];
  const v4f w1 = *(const v4f*)&sw[1][4 * threadIdx.x];
  const v4f w2 = *(const v4f*)&sw[2][4 * threadIdx.x];
  const v4f w3 = *(const v4f*)&sw[3][4 * threadIdx.x];

  for (int tt = 0; tt < TC; tt += 8) {
    // Prefetch this thread's load 24 steps (192 KB) ahead; speculative, safe
    // past the end of the buffer (gfx1250 global_prefetch_b8).
    __builtin_prefetch((const void*)(x4 + idx + (size_t)24 * D4), 0, 0);
#pragma unroll
    for (int u = 0; u < 8; ++u) {
      v4f cur = __builtin_nontemporal_load(&x4[idx]);
      v4f acc = w0 * h0 + w1 * h1 + w2 * h2 + w3 * cur;
      __builtin_nontemporal_store(silu4(acc), &y4[idx]);
      h0 = h1;
      h1 = h2;
      h2 = cur;
      idx += (size_t)D4;
    }
  }
}

extern "C" void kernel_launch(void* const* d_in, const int* in_sizes, int n_in,
                              void* d_out, int out_size, void* d_ws, size_t ws_size,
                              hipStream_t stream) {
  (void)in_sizes; (void)n_in; (void)out_size; (void)d_ws; (void)ws_size;
  const float* x = (const float*)d_in[0];  // (B, T, D) fp32
  const float* w = (const float*)d_in[1];  // (W, 1, D) fp32
  float* y = (float*)d_out;                // (B, T, D) fp32

  dim3 grid(D4 / BLK, T / TC, B);  // (2, 64, 8) = 1024 blocks, 8192 wave32s
  dwconv_silu_kernel<<<grid, dim3(BLK), 0, stream>>>(x, w, y);
}